// DiCNNLink_40621800685767
// MI455X (gfx1250) — compile-verified
//
#include <hip/hip_runtime.h>
#include <hip/hip_bf16.h>

typedef float v2f __attribute__((ext_vector_type(2)));
typedef float v8f __attribute__((ext_vector_type(8)));

#define DEVINL __device__ __forceinline__

DEVINL v8f wmma4(v2f a, v2f b, v8f c) {
    // D = A(16x4) * B(4x16) + C(16x16), all f32, wave32
    return __builtin_amdgcn_wmma_f32_16x16x4_f32(
        /*neg_a=*/false, a, /*neg_b=*/false, b,
        /*c_mod=*/(short)0, c, /*reuse_a=*/false, /*reuse_b=*/false);
}

// A-fragment (16x4, M=time rows, K=channels) from an LDS buffer with row
// stride ST. Lane l (half=l>>4, m=l&15): A.x = buf[r0+m][k0+2*half], A.y next ch.
// Rows clamped at 0 (halo garbage only feeds unused outputs).
DEVINL v2f loadA(const float* buf, int ST, int r0, int kc, int lane) {
    int half = lane >> 4, m = lane & 15;
    int row = r0 + m; if (row < 0) row = 0;
    return *(const v2f*)(buf + row * ST + kc * 4 + 2 * half);
}

// B-fragment (4x16, K x N) for a 1x1 conv weight W[N][K] (row-major, K=Kdim):
// B[k][n] = W[n][k]; lane reads two consecutive K entries of row n0+n.
DEVINL v2f loadB_w(const float* W, int Kdim, int n0, int kc, int lane) {
    int half = lane >> 4, n = lane & 15;
    return *(const v2f*)(W + (n0 + n) * Kdim + kc * 4 + 2 * half);
}

// B-fragment for a K=2 tap conv weight W[N][C][2], fixed tap j.
DEVINL v2f loadB_tap(const float* W, int Cin, int n0, int kc, int j, int lane) {
    int half = lane >> 4, n = lane & 15;
    int c = kc * 4 + 2 * half;
    const float* p = W + ((size_t)(n0 + n) * Cin + c) * 2 + j;
    v2f r; r.x = p[0]; r.y = p[2];
    return r;
}

// Broadcast bias b[n0..n0+15] into all 8 rows of a C fragment.
DEVINL v8f biasC(const float* b, int n0, int lane) {
    float v = b[n0 + (lane & 15)];
    v8f c;
#pragma unroll
    for (int r = 0; r < 8; ++r) c[r] = v;
    return c;
}

// Store / load a 16x16 D/C fragment to/from LDS (stride ST, rows s0.., cols n0..).
DEVINL void storeD(float* buf, int ST, int s0, int n0, v8f d, int lane) {
    int half = lane >> 4, n = lane & 15;
#pragma unroll
    for (int r = 0; r < 8; ++r)
        buf[(s0 + r + 8 * half) * ST + n0 + n] = d[r];
}
DEVINL v8f loadC(const float* buf, int ST, int s0, int n0, int lane) {
    int half = lane >> 4, n = lane & 15;
    v8f c;
#pragma unroll
    for (int r = 0; r < 8; ++r)
        c[r] = buf[(s0 + r + 8 * half) * ST + n0 + n];
    return c;
}

// tanh(p) * sigmoid(p) == (1-u)/(1+u^2) with u = exp(-p): 1 exp + 1 rcp.
DEVINL float gatef(float p) {
    float u = __expf(-p);
    return (1.0f - u) * __builtin_amdgcn_rcpf(1.0f + u * u);
}

// -------- fully fused WaveNet kernel --------
// Chunk of 64 output timesteps per block; 5 waves (160 thr); s = t - t0 + 4.
__global__ __launch_bounds__(160)
void wavenet_fused(const float* __restrict__ x,
                   const float* __restrict__ w_causal, const float* __restrict__ b_causal,
                   const float* __restrict__ wd0, const float* __restrict__ bd0,
                   const float* __restrict__ ws0, const float* __restrict__ bs0,
                   const float* __restrict__ wo0, const float* __restrict__ bo0,
                   const float* __restrict__ wd1, const float* __restrict__ bd1,
                   const float* __restrict__ ws1, const float* __restrict__ bs1,
                   const float* __restrict__ wo1, const float* __restrict__ bo1,
                   const float* __restrict__ w_sk1, const float* __restrict__ b_sk1,
                   const float* __restrict__ w_sk2, const float* __restrict__ b_sk2,
                   float* __restrict__ out)
{
    constexpr int T = 4096, CIN = 64, COUT = 448, HID = 32;
    constexpr int CHUNK = 64;
    constexpr int SDOM  = 80;       // 5 WMMA time-tiles incl. 4-step halo + tail
    constexpr int XROWS = 81;       // x rows se=0..80, t = t0-5+se
    constexpr int XS = 68, ZS = 36; // padded LDS strides (bank-conflict free)

    __shared__ float regionA[2 * SDOM * ZS];   // x (5508) -> later z1 + g (5760)
    __shared__ float z0buf[SDOM * ZS];         // z0 -> later h2
    __shared__ float skipbuf[SDOM * ZS];

    float* xbuf  = regionA;
    float* z1buf = regionA;                    // x dead after stage 2
    float* gbuf  = regionA + SDOM * ZS;
    float* h2buf = z0buf;                      // z0 dead after stage 3

    const int b    = blockIdx.x / (T / CHUNK);
    const int t0   = (blockIdx.x % (T / CHUNK)) * CHUNK;
    const int tid  = threadIdx.x;
    const int lane = tid & 31;
    const int wave = tid >> 5;                 // 0..4
    const int s0   = wave * 16;                // this wave's time-tile

    // ---- Stage 1: async-DMA x[t0-5 .. t0+75] into LDS (zero-fill OOB) ----
    // In-bounds rows: GLOBAL_LOAD_ASYNC_TO_LDS_B128 (ASYNCcnt, bypasses VGPRs).
    // OOB rows: plain zero ds_store (disjoint LDS addresses, so the ASYNCcnt /
    // DScnt ordering split is safe).
    const float* xb = x + (size_t)b * T * CIN;
    const unsigned ldsbase = (unsigned)(size_t)(void*)xbuf;
    for (int idx = tid; idx < XROWS * (CIN / 4); idx += 160) {
        int row = idx >> 4;
        int c4  = (idx & 15) << 2;
        int t   = t0 - 5 + row;
        if (t >= 0 && t < T) {
            unsigned lds_addr = ldsbase + (unsigned)(row * XS + c4) * 4u; // 16B aligned
            unsigned goff     = (unsigned)((t * CIN + c4) * 4);
            asm volatile("global_load_async_to_lds_b128 %0, %1, %2"
                         :: "v"(lds_addr), "v"(goff), "s"(xb) : "memory");
        } else {
            *(float4*)(xbuf + row * XS + c4) = make_float4(0.f, 0.f, 0.f, 0.f);
        }
    }
    asm volatile("s_wait_asynccnt 0" ::: "memory");
    __syncthreads();

    // ---- Stage 2: z0 = causal conv (Cin=64, K=2) ----
#pragma unroll
    for (int n0t = 0; n0t < 2; ++n0t) {
        int n0 = n0t * 16;
        v8f acc = biasC(b_causal, n0, lane);
#pragma unroll
        for (int kc = 0; kc < 16; ++kc) {      // K = 64 channels, step 4
            acc = wmma4(loadA(xbuf, XS, s0,     kc, lane),
                        loadB_tap(w_causal, CIN, n0, kc, 0, lane), acc); // x[t-1]
            acc = wmma4(loadA(xbuf, XS, s0 + 1, kc, lane),
                        loadB_tap(w_causal, CIN, n0, kc, 1, lane), acc); // x[t]
        }
        storeD(z0buf, ZS, s0, n0, acc, lane);
    }
    __syncthreads();   // z0 complete; x dead -> regionA reusable

    // ---- Stage 3: block 0 (dil=1): g0, skip = ws0*g0, z1 = wo0*g0 + z0 ----
#pragma unroll
    for (int n0t = 0; n0t < 2; ++n0t) {
        int n0 = n0t * 16;
        v8f acc = biasC(bd0, n0, lane);
#pragma unroll
        for (int kc = 0; kc < 8; ++kc) {
            acc = wmma4(loadA(z0buf, ZS, s0 - 1, kc, lane),
                        loadB_tap(wd0, HID, n0, kc, 0, lane), acc);
            acc = wmma4(loadA(z0buf, ZS, s0,     kc, lane),
                        loadB_tap(wd0, HID, n0, kc, 1, lane), acc);
        }
#pragma unroll
        for (int r = 0; r < 8; ++r) acc[r] = gatef(acc[r]);
        storeD(gbuf, ZS, s0, n0, acc, lane);   // same-wave LDS: in-order
    }
    {
        v2f ag[8];                              // g0 A-frags, reused 4x
#pragma unroll
        for (int kc = 0; kc < 8; ++kc) ag[kc] = loadA(gbuf, ZS, s0, kc, lane);
#pragma unroll
        for (int n0t = 0; n0t < 2; ++n0t) {
            int n0 = n0t * 16;
            v8f sacc = biasC(bs0, n0, lane);
            v8f zacc = biasC(bo0, n0, lane);
#pragma unroll
            for (int kc = 0; kc < 8; ++kc) {
                sacc = wmma4(ag[kc], loadB_w(ws0, HID, n0, kc, lane), sacc);
                zacc = wmma4(ag[kc], loadB_w(wo0, HID, n0, kc, lane), zacc);
            }
            v8f z0c = loadC(z0buf, ZS, s0, n0, lane);   // residual add
#pragma unroll
            for (int r = 0; r < 8; ++r) zacc[r] += z0c[r];
            storeD(skipbuf, ZS, s0, n0, sacc, lane);
            storeD(z1buf,  ZS, s0, n0, zacc, lane);
        }
    }
    __syncthreads();   // z1 complete (dil=2 reads neighbor rows)

    // ---- Stage 4: block 1 (dil=2): g1, skip += ws1*g1 (residual z2 unused) ----
#pragma unroll
    for (int n0t = 0; n0t < 2; ++n0t) {
        int n0 = n0t * 16;
        v8f acc = biasC(bd1, n0, lane);
#pragma unroll
        for (int kc = 0; kc < 8; ++kc) {
            acc = wmma4(loadA(z1buf, ZS, s0 - 2, kc, lane),
                        loadB_tap(wd1, HID, n0, kc, 0, lane), acc);
            acc = wmma4(loadA(z1buf, ZS, s0,     kc, lane),
                        loadB_tap(wd1, HID, n0, kc, 1, lane), acc);
        }
#pragma unroll
        for (int r = 0; r < 8; ++r) acc[r] = gatef(acc[r]);
        storeD(gbuf, ZS, s0, n0, acc, lane);
    }
    {
        v2f ag[8];
#pragma unroll
        for (int kc = 0; kc < 8; ++kc) ag[kc] = loadA(gbuf, ZS, s0, kc, lane);
#pragma unroll
        for (int n0t = 0; n0t < 2; ++n0t) {
            int n0 = n0t * 16;
            v8f sacc = loadC(skipbuf, ZS, s0, n0, lane);    // accumulate into skip
#pragma unroll
            for (int kc = 0; kc < 8; ++kc)
                sacc = wmma4(ag[kc], loadB_w(ws1, HID, n0, kc, lane), sacc);
            storeD(skipbuf, ZS, s0, n0, sacc, lane);
        }
    }
    __syncthreads();   // skip complete for whole chunk

    // ---- Stage 5: head: relu -> sk1 -> relu -> sk2 (448 out) ----
    if (wave < 4) {
        int so = 4 + 16 * wave;                // output tile rows s in [4,67]
        {
            v2f as[8];                          // relu(skip) A-frags, reused 2x
#pragma unroll
            for (int kc = 0; kc < 8; ++kc) {
                v2f a = loadA(skipbuf, ZS, so, kc, lane);
                a.x = fmaxf(a.x, 0.f); a.y = fmaxf(a.y, 0.f);
                as[kc] = a;
            }
#pragma unroll
            for (int n0t = 0; n0t < 2; ++n0t) {
                int n0 = n0t * 16;
                v8f acc = biasC(b_sk1, n0, lane);
#pragma unroll
                for (int kc = 0; kc < 8; ++kc)
                    acc = wmma4(as[kc], loadB_w(w_sk1, HID, n0, kc, lane), acc);
#pragma unroll
                for (int r = 0; r < 8; ++r) acc[r] = fmaxf(acc[r], 0.f);
                storeD(h2buf, ZS, so, n0, acc, lane);
            }
        }
        // y[t][o] = sk2 * h2 + b_sk2 ; out layout [B, T, COUT]
        float* ob = out + ((size_t)b * T + t0) * COUT;
        const int nh = lane & 15, half = lane >> 4;
        v2f ah[8];                              // h2 A-frags, reused 28x
#pragma unroll
        for (int kc = 0; kc < 8; ++kc) ah[kc] = loadA(h2buf, ZS, so, kc, lane);
#pragma unroll 2
        for (int n0 = 0; n0 < COUT; n0 += 16) {
            v8f acc = biasC(b_sk2, n0, lane);
#pragma unroll
            for (int kc = 0; kc < 8; ++kc)
                acc = wmma4(ah[kc], loadB_w(w_sk2, HID, n0, kc, lane), acc);
#pragma unroll
            for (int r = 0; r < 8; ++r) {
                int s = so + r + 8 * half;                     // t = t0 + s - 4
                ob[(size_t)(s - 4) * COUT + n0 + nh] = acc[r];
            }
        }
    }
}

extern "C" void kernel_launch(void* const* d_in, const int* in_sizes, int n_in,
                              void* d_out, int out_size, void* d_ws, size_t ws_size,
                              hipStream_t stream) {
    (void)in_sizes; (void)n_in; (void)out_size; (void)d_ws; (void)ws_size;
    const float* x        = (const float*)d_in[0];
    const float* w_causal = (const float*)d_in[1];
    const float* b_causal = (const float*)d_in[2];
    const float* wd0 = (const float*)d_in[3];
    const float* bd0 = (const float*)d_in[4];
    const float* ws0 = (const float*)d_in[5];
    const float* bs0 = (const float*)d_in[6];
    const float* wo0 = (const float*)d_in[7];
    const float* bo0 = (const float*)d_in[8];
    const float* wd1 = (const float*)d_in[9];
    const float* bd1 = (const float*)d_in[10];
    const float* ws1 = (const float*)d_in[11];
    const float* bs1 = (const float*)d_in[12];
    const float* wo1 = (const float*)d_in[13];
    const float* bo1 = (const float*)d_in[14];
    const float* w_sk1 = (const float*)d_in[15];
    const float* b_sk1 = (const float*)d_in[16];
    const float* w_sk2 = (const float*)d_in[17];
    const float* b_sk2 = (const float*)d_in[18];
    float* out = (float*)d_out;

    dim3 grid(32 * (4096 / 64));   // batch x time-chunks = 2048 blocks
    dim3 block(160);               // 5 wave32s
    wavenet_fused<<<grid, block, 0, stream>>>(
        x, w_causal, b_causal,
        wd0, bd0, ws0, bs0, wo0, bo0,
        wd1, bd1, ws1, bs1, wo1, bo1,
        w_sk1, b_sk1, w_sk2, b_sk2, out);
}